// FCOSPostprocessor_78168404787575
// MI455X (gfx1250) — compile-verified
//
#include <hip/hip_runtime.h>
#include <hip/hip_bf16.h>
#include <math.h>
#include <stdint.h>

#ifndef __has_builtin
#define __has_builtin(x) 0
#endif

#if defined(__HIP_DEVICE_COMPILE__) && __has_builtin(__builtin_amdgcn_tensor_load_to_lds)
#define USE_TDM 1
#endif
#if defined(__HIP_DEVICE_COMPILE__) && __has_builtin(__builtin_amdgcn_s_wait_tensorcnt)
#define HAVE_WAIT_TENSOR 1
#endif
#if defined(__HIP_DEVICE_COMPILE__) && __has_builtin(__builtin_amdgcn_global_load_async_to_lds_b32) && __has_builtin(__builtin_amdgcn_s_wait_asynccnt)
#define USE_ASYNC 1
#endif
#if defined(__HIP_DEVICE_COMPILE__) && __has_builtin(__builtin_amdgcn_wmma_f32_16x16x4_f32)
#define USE_WMMA_SCAN 1
#endif

// Hard probe: on the gfx1250 device pass we REQUIRE the TDM builtin so the
// compile result proves tensor_load_to_lds is in the binary. (Confirmed round 2.)
#if defined(__HIP_DEVICE_COMPILE__) && defined(__gfx1250__) && !defined(USE_TDM)
#error "gfx1250: __builtin_amdgcn_tensor_load_to_lds unavailable on this toolchain"
#endif

#define NLEV  5
#define NBATCH 16
#define NCLS  80
#define TOPN  300
#define KCAND 1500      // 5 * TOPN
#define POSTN 100
#define NBINS 2048
#define NTASK (NBATCH * NLEV)
#define KP    2048      // sort padding
#define SOA   (KCAND * 6)   // per-image candidate SoA floats

// H*W per level for IMG 800x1024, strides 8/16/32/64/128
__device__ __constant__ int c_HW[NLEV] = {12800, 3200, 800, 208, 56};

typedef float v2f __attribute__((ext_vector_type(2)));
typedef float v8f __attribute__((ext_vector_type(8)));

struct InPtrs {
  const float* loc[NLEV];
  const float* cls[NLEV];
  const float* box[NLEV];
  const float* ctr[NLEV];
  const int* ih;
  const int* iw;
};

__device__ __forceinline__ unsigned fkey(float f) {
  unsigned u = __float_as_uint(f);
  return (u & 0x80000000u) ? ~u : (u | 0x80000000u);  // monotonic: bigger float -> bigger key
}
__device__ __forceinline__ float sigmoidf(float x) { return 1.0f / (1.0f + expf(-x)); }

// ---------------- radix-select pass: histogram of key bits ----------------
__global__ void hist_kernel(InPtrs in, unsigned* hist, unsigned* state,
                            int pass, int shift, int bits) {
  int task = blockIdx.y;
  int b = task / NLEV, li = task % NLEV;
  int HW = c_HW[li];
  int N = HW * NCLS;
  __shared__ unsigned lh[NBINS];
  for (int i = threadIdx.x; i < NBINS; i += blockDim.x) lh[i] = 0u;
  __syncthreads();
  const float* cls = in.cls[li] + (size_t)b * (size_t)N;
  const float* ctr = in.ctr[li] + (size_t)b * (size_t)HW;
  unsigned prefix = (pass > 0) ? state[task * 4 + 0] : 0u;
  int shiftAbove = shift + bits;
  unsigned mask = (1u << bits) - 1u;
  int stride = gridDim.x * blockDim.x;
  for (int j = blockIdx.x * blockDim.x + threadIdx.x; j < N; j += stride) {
    __builtin_prefetch(cls + j + stride, 0, 1);   // -> global_prefetch_b8
    float x = cls[j];
    int c = (int)((unsigned)j / (unsigned)HW);
    int pos = j - c * HW;
    float sc = sigmoidf(x);
    float v = (sc > 0.05f) ? sc * sigmoidf(ctr[pos]) : -1.0f;
    unsigned key = fkey(v);
    if (pass == 0 || ((key ^ prefix) >> shiftAbove) == 0u)
      atomicAdd(&lh[(key >> shift) & mask], 1u);
  }
  __syncthreads();
  int nb = 1 << bits;
  for (int i = threadIdx.x; i < nb; i += blockDim.x)
    if (lh[i]) atomicAdd(&hist[(size_t)task * NBINS + i], lh[i]);
}

// ---------------- pick the bin containing the 300th largest ----------------
// Segment totals (16 bins each) are computed with V_WMMA_F32_16X16X4_F32 using a
// ones A-matrix: D = ones(16x4) * B(4x16) + C  => every row of D = column sums.
// Four accumulating WMMAs reduce a 256-bin chunk to 16 segment totals (exact:
// integer counts < 2^24). Thread 0 then descends segments -> bins.
__global__ void scan_kernel(unsigned* hist, unsigned* state, int pass, int shift, int bits) {
  int task = blockIdx.x;
  unsigned* h = hist + (size_t)task * NBINS;
  __shared__ unsigned sh[NBINS];
  __shared__ float segsum[NBINS / 16];
#ifdef USE_ASYNC
  // per-lane async VMEM->LDS staging of the histogram (global_load_async_to_lds_b32)
  for (int i = threadIdx.x; i < NBINS; i += blockDim.x) {
    __builtin_amdgcn_global_load_async_to_lds_b32(
        (__attribute__((address_space(1))) int*)(h + i),
        (__attribute__((address_space(3))) int*)(sh + i), 0, 0);
  }
  __builtin_amdgcn_s_wait_asynccnt(0);
#else
  for (int i = threadIdx.x; i < NBINS; i += blockDim.x) sh[i] = h[i];
#endif
  __syncthreads();
  int nb = 1 << bits;
#ifdef USE_WMMA_SCAN
  if (threadIdx.x < 32) {            // wave 0, EXEC all ones
    int lane = threadIdx.x;
    int col = lane & 15;
    int bk = (lane >> 4) ? 2 : 0;    // B VGPR0 holds rows K={0,2}, VGPR1 rows K={1,3}
    int nchunk = nb / 256;
    v2f a; a.x = 1.0f; a.y = 1.0f;   // ones A-matrix
    for (int ch = 0; ch < nchunk; ++ch) {
      v8f acc = {};
      for (int k4 = 0; k4 < 4; ++k4) {
        int base = ch * 256 + col * 16 + k4 * 4;
        v2f bv;
        bv.x = (float)sh[base + bk];
        bv.y = (float)sh[base + bk + 1];
        acc = __builtin_amdgcn_wmma_f32_16x16x4_f32(false, a, false, bv,
                                                    (short)0, acc, false, false);
      }
      if (lane < 16) segsum[ch * 16 + col] = acc[0];  // D row M=0
    }
  }
  __syncthreads();
  if (threadIdx.x == 0) {
    unsigned need = (pass == 0) ? (unsigned)TOPN : state[task * 4 + 1];
    unsigned prefix = (pass == 0) ? 0u : state[task * 4 + 0];
    int nseg = nb / 16;
    unsigned acc = 0; int sel = 0;
    for (int s = nseg - 1; s >= 0; --s) {
      unsigned cs = (unsigned)segsum[s];
      if (acc + cs >= need) {
        for (int bin = s * 16 + 15; bin >= s * 16; --bin) {
          unsigned c = sh[bin];
          if (acc + c >= need) { sel = bin; need -= acc; break; }
          acc += c;
        }
        break;
      }
      acc += cs;
    }
    prefix |= ((unsigned)sel) << shift;
    state[task * 4 + 0] = prefix;     // after pass 2: exact key T of 300th largest
    state[task * 4 + 1] = need;       // #elements == T still required
    if (pass == 2) { state[task * 4 + 2] = 0u; state[task * 4 + 3] = 0u; }
  }
#else
  (void)segsum;
  if (threadIdx.x == 0) {
    unsigned need = (pass == 0) ? (unsigned)TOPN : state[task * 4 + 1];
    unsigned prefix = (pass == 0) ? 0u : state[task * 4 + 0];
    unsigned acc = 0; int sel = 0;
    for (int bin = nb - 1; bin >= 0; --bin) {
      unsigned c = sh[bin];
      if (acc + c >= need) { sel = bin; need -= acc; break; }
      acc += c;
    }
    prefix |= ((unsigned)sel) << shift;
    state[task * 4 + 0] = prefix;
    state[task * 4 + 1] = need;
    if (pass == 2) { state[task * 4 + 2] = 0u; state[task * 4 + 3] = 0u; }
  }
#endif
  __syncthreads();
  for (int i = threadIdx.x; i < NBINS; i += blockDim.x) h[i] = 0u;  // ready for next pass
}

// ---------------- gather the exact top-300 (value, flat idx) ----------------
__global__ void collect_kernel(InPtrs in, unsigned* state, float* cval, int* cidx) {
  int task = blockIdx.y;
  int b = task / NLEV, li = task % NLEV;
  int HW = c_HW[li];
  int N = HW * NCLS;
  const float* cls = in.cls[li] + (size_t)b * (size_t)N;
  const float* ctr = in.ctr[li] + (size_t)b * (size_t)HW;
  unsigned T = state[task * 4 + 0];
  unsigned need = state[task * 4 + 1];
  int stride = gridDim.x * blockDim.x;
  for (int j = blockIdx.x * blockDim.x + threadIdx.x; j < N; j += stride) {
    __builtin_prefetch(cls + j + stride, 0, 1);
    float x = cls[j];
    int c = (int)((unsigned)j / (unsigned)HW);
    int pos = j - c * HW;
    float sc = sigmoidf(x);
    float v = (sc > 0.05f) ? sc * sigmoidf(ctr[pos]) : -1.0f;
    unsigned key = fkey(v);
    int slot = -1;
    if (key > T) {
      slot = (int)atomicAdd(&state[task * 4 + 2], 1u);       // exactly 300-need of these
    } else if (key == T) {
      unsigned e = atomicAdd(&state[task * 4 + 3], 1u);
      if (e < need) slot = (int)((unsigned)TOPN - need + e);
    }
    if (slot >= 0) {
      cval[task * TOPN + slot] = v;
      cidx[task * TOPN + slot] = pos * NCLS + c;  // flat idx in (HW, C) order
    }
  }
}

// ---------------- decode boxes, build per-image candidate SoA ----------------
__global__ void decode_kernel(InPtrs in, const float* cval, const int* cidx, float* cands) {
  int task = blockIdx.x;
  int n = threadIdx.x;
  if (n >= TOPN) return;
  int b = task / NLEV, li = task % NLEV;
  int HW = c_HW[li];
  float v = cval[task * TOPN + n];
  int idx = cidx[task * TOPN + n];
  int pos = (int)((unsigned)idx / (unsigned)NCLS);
  int cl = idx - pos * NCLS;
  int label = cl + 1;
  const float* box = in.box[li] + (size_t)b * 4u * (size_t)HW;
  const float* loc = in.loc[li];
  float lx = loc[2 * pos], ly = loc[2 * pos + 1];
  float b0 = box[pos], b1 = box[HW + pos], b2 = box[2 * HW + pos], b3 = box[3 * HW + pos];
  float fw = (float)(*in.iw) - 1.0f;
  float fh = (float)(*in.ih) - 1.0f;
  float x1 = fminf(fmaxf(lx - b0, 0.0f), fw);
  float y1 = fminf(fmaxf(ly - b1, 0.0f), fh);
  float x2 = fminf(fmaxf(lx + b2, 0.0f), fw);
  float y2 = fminf(fmaxf(ly + b3, 0.0f), fh);
  bool valid = (v > 0.0f) && ((x2 - x1 + 1.0f) >= 0.0f) && ((y2 - y1 + 1.0f) >= 0.0f);
  float score = valid ? sqrtf(fmaxf(v, 1e-12f)) : -INFINITY;
  float* ib = cands + (size_t)b * (size_t)SOA;
  int q = li * TOPN + n;
  ib[q] = x1;                       // x1 @ 0
  ib[KCAND + q] = y1;               // y1 @ 1500
  ib[2 * KCAND + q] = x2;           // x2 @ 3000
  ib[3 * KCAND + q] = y2;           // y2 @ 4500
  ib[4 * KCAND + q] = score;        // score @ 6000
  ib[5 * KCAND + q] = (float)label; // label @ 7500
}

// ---- per-image NMS: TDM DMA of full 36KB candidate SoA into LDS, bitonic sort,
// ---- greedy suppression fully out of LDS
__global__ __launch_bounds__(1024) void nms_kernel(const float* cands, const int* ihp,
                                                   const int* iwp, float* out) {
  int b = blockIdx.x;
  int tid = threadIdx.x;
  int bs = blockDim.x;
  __shared__ float stage[SOA];     // 36000 B: x1|y1|x2|y2|score|label
  __shared__ float skey[KP];
  __shared__ int sidx[KP];
  __shared__ unsigned char kflag[KCAND];

  const float* ib = cands + (size_t)b * (size_t)SOA;

#ifdef USE_TDM
  {
    typedef unsigned int tdm_u4 __attribute__((ext_vector_type(4)));
    typedef int tdm_i8 __attribute__((ext_vector_type(8)));
    typedef int tdm_i4 __attribute__((ext_vector_type(4)));
    if (tid == 0) {
      // One D#: 1 row x 9000 dwords, global -> LDS.
      uint64_t ga = (uint64_t)(uintptr_t)ib;
      uint32_t la = (uint32_t)(uintptr_t)(void*)&stage[0];
      tdm_u4 g0;
      g0[0] = 1u;                                            // count=1, user descriptor
      g0[1] = la;                                            // lds_addr (bytes)
      g0[2] = (uint32_t)(ga & 0xFFFFFFFFull);                // global_addr[31:0]
      g0[3] = (uint32_t)((ga >> 32) & 0x01FFFFFFull) | (2u << 30);  // addr hi | type=2
      tdm_i8 g1;
      g1[0] = (int)(2u << 16);                               // data_size = 4B, wg_mask=0
      g1[1] = (int)((unsigned)SOA << 16);                    // tensor_dim0[15:0] = 9000
      g1[2] = (int)(1u << 16);                               // dim0 hi=0 | tensor_dim1 lo=1
      g1[3] = (int)((unsigned)SOA << 16);                    // dim1 hi=0 | tile_dim0=9000
      g1[4] = 1;                                             // tile_dim1=1, tile_dim2=0
      g1[5] = SOA;                                           // tensor_dim0_stride lo
      g1[6] = (int)((unsigned)SOA << 16);                    // stride0 hi=0 | stride1 lo
      g1[7] = 0;
      tdm_i4 g2; g2[0] = 1; g2[1] = 1; g2[2] = 0; g2[3] = 0; // dim2=1, dim3=1, tile_dim3=0
      tdm_i4 g3; g3[0] = 0; g3[1] = 0; g3[2] = 0; g3[3] = 0;
#if __clang_major__ >= 23
      tdm_i8 g4; g4[0]=0; g4[1]=0; g4[2]=0; g4[3]=0; g4[4]=0; g4[5]=0; g4[6]=0; g4[7]=0;
      __builtin_amdgcn_tensor_load_to_lds(g0, g1, g2, g3, g4, 0);
#else
      __builtin_amdgcn_tensor_load_to_lds(g0, g1, g2, g3, 0);
#endif
    }
#ifdef HAVE_WAIT_TENSOR
    __builtin_amdgcn_s_wait_tensorcnt(0);
#endif
  }
#else
  for (int i = tid; i < SOA; i += bs) stage[i] = ib[i];
#endif
  __syncthreads();

  for (int i = tid; i < KP; i += bs) {
    skey[i] = (i < KCAND) ? stage[4 * KCAND + i] : -INFINITY;
    sidx[i] = i;
  }

  // bitonic sort, descending by score
  for (int kk = 2; kk <= KP; kk <<= 1) {
    for (int j = kk >> 1; j > 0; j >>= 1) {
      __syncthreads();
      for (int i = tid; i < KP; i += bs) {
        int ixj = i ^ j;
        if (ixj > i) {
          float a = skey[i], c = skey[ixj];
          bool up = ((i & kk) == 0);
          if (up ? (a < c) : (a > c)) {
            skey[i] = c; skey[ixj] = a;
            int t = sidx[i]; sidx[i] = sidx[ixj]; sidx[ixj] = t;
          }
        }
      }
    }
  }
  __syncthreads();

  for (int i = tid; i < KCAND; i += bs)
    kflag[i] = (skey[i] > 0.0f) ? (unsigned char)1 : (unsigned char)0;
  int mh = *ihp, mw = *iwp;
  float off = (float)((mh > mw ? mh : mw) + 1);
  __syncthreads();

  // greedy NMS in sorted order; class-offset only affects the min/max corners
  for (int i = 0; i < KCAND; ++i) {
    if (kflag[i]) {
      int gi = sidx[i];
      float oi = off * stage[5 * KCAND + gi];
      float ax1 = stage[gi] + oi, ay1 = stage[KCAND + gi] + oi;
      float ax2 = stage[2 * KCAND + gi] + oi, ay2 = stage[3 * KCAND + gi] + oi;
      float aarea = (stage[2 * KCAND + gi] - stage[gi]) *
                    (stage[3 * KCAND + gi] - stage[KCAND + gi]);
      for (int j = i + 1 + tid; j < KCAND; j += bs) {
        if (kflag[j]) {
          int gj = sidx[j];
          float oj = off * stage[5 * KCAND + gj];
          float jx1 = stage[gj], jy1 = stage[KCAND + gj];
          float jx2 = stage[2 * KCAND + gj], jy2 = stage[3 * KCAND + gj];
          float ix1 = fmaxf(ax1, jx1 + oj);
          float iy1 = fmaxf(ay1, jy1 + oj);
          float ix2 = fminf(ax2, jx2 + oj);
          float iy2 = fminf(ay2, jy2 + oj);
          float w = fmaxf(ix2 - ix1, 0.0f);
          float h = fmaxf(iy2 - iy1, 0.0f);
          float inter = w * h;
          float ba = (jx2 - jx1) * (jy2 - jy1);
          float uni = fmaxf(aarea + ba - inter, 1e-9f);
          if (inter / uni > 0.6f) kflag[j] = 0;
        }
      }
    }
    __syncthreads();
  }

  // outputs: out5 [B,100,5] | labels [B,100] | valid [B,100], all as f32
  float* out5 = out;
  float* outl = out + NBATCH * POSTN * 5;
  float* outv = out + NBATCH * POSTN * 5 + NBATCH * POSTN;
  for (int t = tid; t < POSTN * 7; t += bs) {
    if (t < POSTN * 5) out5[b * POSTN * 5 + t] = 0.0f;
    else if (t < POSTN * 6) outl[b * POSTN + (t - POSTN * 5)] = 0.0f;
    else outv[b * POSTN + (t - POSTN * 6)] = 0.0f;
  }
  __syncthreads();
  if (tid == 0) {
    int cnt = 0;
    for (int i = 0; i < KCAND && cnt < POSTN; ++i) {
      if (kflag[i]) {
        int gi = sidx[i];
        float* o = out5 + (size_t)b * POSTN * 5 + cnt * 5;
        o[0] = stage[gi];
        o[1] = stage[KCAND + gi];
        o[2] = stage[2 * KCAND + gi];
        o[3] = stage[3 * KCAND + gi];
        o[4] = skey[i];
        outl[b * POSTN + cnt] = stage[5 * KCAND + gi];
        outv[b * POSTN + cnt] = 1.0f;
        ++cnt;
      }
    }
  }
}

extern "C" void kernel_launch(void* const* d_in, const int* in_sizes, int n_in,
                              void* d_out, int out_size, void* d_ws, size_t ws_size,
                              hipStream_t stream) {
  (void)in_sizes; (void)n_in; (void)out_size; (void)ws_size;
  InPtrs P;
  for (int li = 0; li < NLEV; ++li) {
    P.loc[li] = (const float*)d_in[4 * li + 0];
    P.cls[li] = (const float*)d_in[4 * li + 1];
    P.box[li] = (const float*)d_in[4 * li + 2];
    P.ctr[li] = (const float*)d_in[4 * li + 3];
  }
  P.ih = (const int*)d_in[20];
  P.iw = (const int*)d_in[21];

  // workspace layout (bytes)
  char* ws = (char*)d_ws;
  size_t offHist = 0;                                        // 80*2048*4 = 655360
  size_t offState = offHist + (size_t)NTASK * NBINS * 4;     // 80*16    = 1280
  size_t offCVal = offState + (size_t)NTASK * 16;            // 80*300*4 = 96000
  size_t offCIdx = offCVal + (size_t)NTASK * TOPN * 4;       // 96000
  size_t offCand = offCIdx + (size_t)NTASK * TOPN * 4;       // 16*9000*4 = 576000
  unsigned* hist = (unsigned*)(ws + offHist);
  unsigned* state = (unsigned*)(ws + offState);
  float* cval = (float*)(ws + offCVal);
  int* cidx = (int*)(ws + offCIdx);
  float* cands = (float*)(ws + offCand);

  (void)hipMemsetAsync(hist, 0, (size_t)NTASK * NBINS * 4 + (size_t)NTASK * 16, stream);

  dim3 gh(64, NTASK);
  // 3-pass radix select: 11 + 11 + 10 bits of the monotonic float key
  hist_kernel<<<gh, 256, 0, stream>>>(P, hist, state, 0, 21, 11);
  scan_kernel<<<NTASK, 256, 0, stream>>>(hist, state, 0, 21, 11);
  hist_kernel<<<gh, 256, 0, stream>>>(P, hist, state, 1, 10, 11);
  scan_kernel<<<NTASK, 256, 0, stream>>>(hist, state, 1, 10, 11);
  hist_kernel<<<gh, 256, 0, stream>>>(P, hist, state, 2, 0, 10);
  scan_kernel<<<NTASK, 256, 0, stream>>>(hist, state, 2, 0, 10);
  collect_kernel<<<gh, 256, 0, stream>>>(P, state, cval, cidx);
  decode_kernel<<<NTASK, 320, 0, stream>>>(P, cval, cidx, cands);
  nms_kernel<<<NBATCH, 1024, 0, stream>>>(cands, P.ih, P.iw, (float*)d_out);
}